// Attn_40046275068166
// MI455X (gfx1250) — compile-verified
//
#include <hip/hip_runtime.h>

typedef float v2f __attribute__((ext_vector_type(2)));
typedef float v4f __attribute__((ext_vector_type(4)));
typedef float v8f __attribute__((ext_vector_type(8)));

#define TT   34          // tokens per batch
#define DM   6           // d_model
#define NB   8           // batches per workgroup
#define NHD  2           // heads
#define HD   3           // head dim
#define NTHREADS 256
#define NWAVE (NTHREADS/32)
#define NTOK (NB*TT)     // 272 token-slots per block
#define NTILE (NTOK/16)  // 17 WMMA tiles (exact)
#define KVB  (TT*NHD*8 + 8)   // per-batch kv stride in floats (552), 16B-multiple, bank-skewed

// smem layout (floats):
//  [0,     4352)  s_qk : col-major 16 x NTOK (phase1: cols0-5=q, 6-11=k ; phase3: out)
//  [4352,  8704)  s_vv : col-major 16 x NTOK (cols0-5=v)
//  [8704, 10336)  s_x  (phase 0/1)  ALIASED WITH  s_o (phase 2/3)  -- disjoint lifetimes
//  [10336,14752)  s_kv : packed {k0,k1,k2,_,v0,v1,v2,_} per (token,head)
#define SMEM_TOTAL 14752

__global__ __launch_bounds__(NTHREADS)
void attn_tiny_kernel(const float* __restrict__ x,
                      const float* __restrict__ Wq,
                      const float* __restrict__ Wk,
                      const float* __restrict__ Wv,
                      const float* __restrict__ Wo,
                      float* __restrict__ out,
                      int Btot)
{
    __shared__ float smem[SMEM_TOTAL];
    float* const s_qk = smem;
    float* const s_vv = smem + 4352;
    float* const s_x  = smem + 8704;
    float* const s_o  = smem + 8704;    // alias: s_x dead after phase 1
    float* const s_kv = smem + 10336;

    const int tid  = threadIdx.x;
    const int lane = tid & 31;
    const int wave = tid >> 5;
    const int hi   = lane >> 4;         // 0 = lanes 0-15, 1 = lanes 16-31
    const int lx   = lane & 15;

    const long long elemBase   = (long long)blockIdx.x * NTOK * DM;
    const long long totalElems = (long long)Btot * TT * DM;

    // ---- coalesced stage of x tile into LDS ----
    for (int i = tid; i < NTOK*DM; i += NTHREADS) {
        long long g = elemBase + i;
        s_x[i] = (g < totalElems) ? x[g] : 0.f;
    }

    // ---- per-lane constant WMMA B-matrices (weights, tile-invariant) ----
    // B layout (4x16, KxN): VGPR0: lanes0-15 K=0 / lanes16-31 K=2,
    //                       VGPR1: lanes0-15 K=1 / lanes16-31 K=3
    v2f bqk, bv, bo1, bo2;
    {
        float w0=0.f,w1=0.f,w2=0.f;
        if      (lx < 6)  { w0=Wq[lx*3+0];       w1=Wq[lx*3+1];       w2=Wq[lx*3+2]; }
        else if (lx < 12) { int n=lx-6; w0=Wk[n*3+0]; w1=Wk[n*3+1]; w2=Wk[n*3+2]; }
        bqk.x = hi ? w2  : w0;
        bqk.y = hi ? 0.f : w1;

        float u0=0.f,u1=0.f,u2=0.f;
        if (lx < 6) { u0=Wv[lx*3+0]; u1=Wv[lx*3+1]; u2=Wv[lx*3+2]; }
        bv.x = hi ? u2  : u0;
        bv.y = hi ? 0.f : u1;

        float o0=0.f,o1=0.f,o2=0.f,o3=0.f,o4=0.f,o5=0.f;
        if (lx < 6) { o0=Wo[lx*6+0]; o1=Wo[lx*6+1]; o2=Wo[lx*6+2];
                      o3=Wo[lx*6+3]; o4=Wo[lx*6+4]; o5=Wo[lx*6+5]; }
        bo1.x = hi ? o2  : o0;
        bo1.y = hi ? o3  : o1;
        bo2.x = hi ? 0.f : o4;
        bo2.y = hi ? 0.f : o5;
    }

    __syncthreads();

    // ---- phase 1: Q,K,V projections via V_WMMA_F32_16X16X4_F32 ----
    for (int tile = wave; tile < NTILE; tile += NWAVE) {
        const int slot   = tile*16 + lx;
        const int xb     = slot*DM;
        const int cmBase = lx*NTOK + tile*16 + hi*8;

        v2f a;                               // xp = x[..., 3:6], K padded to 4
        a.x = s_x[xb + 3 + 2*hi];
        float ay = s_x[xb + 4];
        a.y = hi ? 0.f : ay;
        v8f c = {};
        c = __builtin_amdgcn_wmma_f32_16x16x4_f32(false, a, false, bqk,
                                                  (short)0, c, false, false);
        *(v4f*)&s_qk[cmBase]     = (v4f){c[0], c[1], c[2], c[3]};
        *(v4f*)&s_qk[cmBase + 4] = (v4f){c[4], c[5], c[6], c[7]};

        v2f av;                              // xt = x[..., 0:3]
        av.x = s_x[xb + 2*hi];
        float avy = s_x[xb + 1];
        av.y = hi ? 0.f : avy;
        v8f cv = {};
        cv = __builtin_amdgcn_wmma_f32_16x16x4_f32(false, av, false, bv,
                                                   (short)0, cv, false, false);
        *(v4f*)&s_vv[cmBase]     = (v4f){cv[0], cv[1], cv[2], cv[3]};
        *(v4f*)&s_vv[cmBase + 4] = (v4f){cv[4], cv[5], cv[6], cv[7]};
    }

    __syncthreads();

    // ---- phase 1.5: repack k,v into {k0,k1,k2,_,v0,v1,v2,_} per (token,head) ----
    for (int i = tid; i < NTOK*NHD; i += NTHREADS) {
        int bt   = i >> 1;
        int h    = i & 1;
        int b    = bt / TT;
        int tloc = bt - b*TT;
        float k0 = s_qk[(6 + h*HD + 0)*NTOK + bt];
        float k1 = s_qk[(6 + h*HD + 1)*NTOK + bt];
        float k2 = s_qk[(6 + h*HD + 2)*NTOK + bt];
        float u0 = s_vv[(h*HD + 0)*NTOK + bt];
        float u1 = s_vv[(h*HD + 1)*NTOK + bt];
        float u2 = s_vv[(h*HD + 2)*NTOK + bt];
        float* d = &s_kv[b*KVB + (tloc*NHD + h)*8];
        *(v4f*)(d)     = (v4f){k0, k1, k2, 0.f};
        *(v4f*)(d + 4) = (v4f){u0, u1, u2, 0.f};
    }

    __syncthreads();

    // ---- phase 2: causal softmax attention (max-free, exp2 domain) ----
    // tasks ordered t-major so each wave's 32 lanes share (t, t+1) -> minimal divergence
    const float SCL = 0.57735026918962576f * 1.4426950408889634f;  // (1/sqrt(3))*log2(e)
    for (int task = tid; task < NTOK*NHD; task += NTHREADS) {
        int t  = task >> 4;          // 0..33
        int bh = task & 15;
        int b  = bh >> 1;
        int h  = bh & 1;
        int bt = b*TT + t;

        float q0 = s_qk[(h*HD + 0)*NTOK + bt] * SCL;
        float q1 = s_qk[(h*HD + 1)*NTOK + bt] * SCL;
        float q2 = s_qk[(h*HD + 2)*NTOK + bt] * SCL;

        const float* kv = &s_kv[b*KVB + h*8];
        float l = 0.f, a0 = 0.f, a1 = 0.f, a2 = 0.f;
        for (int j = 0; j <= t; ++j) {
            v4f kk = *(const v4f*)(kv + j*16);
            v4f vv = *(const v4f*)(kv + j*16 + 4);
            float p = __builtin_amdgcn_exp2f(q0*kk.x + q1*kk.y + q2*kk.z);
            l  += p;
            a0 += p*vv.x;
            a1 += p*vv.y;
            a2 += p*vv.z;
        }
        float inv = __builtin_amdgcn_rcpf(l);
        float* op = &s_o[bt*DM + h*HD];
        op[0] = a0*inv; op[1] = a1*inv; op[2] = a2*inv;
    }

    __syncthreads();

    // ---- phase 3: output projection (K=6 split as 4+2, C-chained WMMAs) ----
    for (int tile = wave; tile < NTILE; tile += NWAVE) {
        const int slot   = tile*16 + lx;
        const int ob     = slot*DM;
        const int cmBase = lx*NTOK + tile*16 + hi*8;

        v2f a1v, a2v;
        a1v.x = s_o[ob + 2*hi];
        a1v.y = s_o[ob + 1 + 2*hi];
        float t4 = s_o[ob + 4];
        float t5 = s_o[ob + 5];
        a2v.x = hi ? 0.f : t4;
        a2v.y = hi ? 0.f : t5;

        v8f c = {};
        c = __builtin_amdgcn_wmma_f32_16x16x4_f32(false, a1v, false, bo1,
                                                  (short)0, c, false, false);
        c = __builtin_amdgcn_wmma_f32_16x16x4_f32(false, a2v, false, bo2,
                                                  (short)0, c, false, false);
        *(v4f*)&s_qk[cmBase]     = (v4f){c[0], c[1], c[2], c[3]};
        *(v4f*)&s_qk[cmBase + 4] = (v4f){c[4], c[5], c[6], c[7]};
    }

    __syncthreads();

    // ---- coalesced global store of the projected output ----
    for (int i = tid; i < NTOK*DM; i += NTHREADS) {
        long long g = elemBase + i;
        if (g < totalElems) {
            int row = i / DM;
            int col = i - row*DM;
            out[g] = s_qk[col*NTOK + row];
        }
    }
}

extern "C" void kernel_launch(void* const* d_in, const int* in_sizes, int n_in,
                              void* d_out, int out_size, void* d_ws, size_t ws_size,
                              hipStream_t stream) {
    const float* x  = (const float*)d_in[0];
    const float* Wq = (const float*)d_in[1];
    const float* Wk = (const float*)d_in[2];
    const float* Wv = (const float*)d_in[3];
    const float* Wo = (const float*)d_in[4];
    float* out = (float*)d_out;

    int Btot = in_sizes[0] / (TT * DM);
    int grid = (Btot + NB - 1) / NB;
    attn_tiny_kernel<<<grid, NTHREADS, 0, stream>>>(x, Wq, Wk, Wv, Wo, out, Btot);
}